// GeometricAttentionHead_43052752175153
// MI455X (gfx1250) — compile-verified
//
#include <hip/hip_runtime.h>

// ---------------------------------------------------------------------------
// GeometricAttentionHead for MI455X (gfx1250, wave32, WMMA)
//   out  = softmax(-acosh(clip(-<x,x>_L)) / temp) @ values
// d_out = [ output : B*S*D floats | attention : B*S*S floats ]
//
// Key fact: dist = acosh(clip(u, 1+1e-7, 50)) in [4.5e-4, 4.61], so
// scores <= 0 and exp(score) in (0,1]: softmax needs NO max-subtraction
// pass (shift-invariant, no overflow possible). Pipeline touches the 134MB
// attention matrix only 3x (HBM-bound optimum ~402MB @ 23.3TB/s ~ 17us):
//   K0: zero per-row sum scratch (d_ws)
//   K1: fp32 WMMA Gram -> p=exp(score), write p, atomic per-row sums
//   K3: normalize-in-place while staging + fp32 WMMA  attention @ values
// x (4.3MB) and values (532KB/batch) stay L2-resident (192MB L2) across
// their heavy re-reads; only the attention stream touches HBM.
// ---------------------------------------------------------------------------

typedef __attribute__((ext_vector_type(2))) float v2f;
typedef __attribute__((ext_vector_type(8))) float v8f;

constexpr int   BB        = 8;
constexpr int   SS        = 2048;
constexpr int   DD        = 65;
constexpr int   KPAD      = 68;      // D padded to mult. of 4 (LDS stride too)
constexpr float ACOSH_MIN = 1.0f + 1e-07f;
constexpr float ACOSH_MAX = 50.0f;
constexpr float TEMP_EPS  = 1e-08f;

// -------------------------------------------------------------------------
__global__ __launch_bounds__(256) void zero_rowsums_kernel(float* __restrict__ rs)
{
    const int i = blockIdx.x * 256 + threadIdx.x;
    if (i < BB * SS) rs[i] = 0.0f;
}

// -------------------------------------------------------------------------
// Kernel 1: 64x64 tile of p = exp(-acosh(clip(-<x,x>_L))/temp) per block.
// 8 waves (2x4), two 16x16 fp32-WMMA accumulators per wave, K padded to 68.
// Lorentz signature folded into A (negated time component). Epilogue also
// reduces per-row partial sums: shfl_xor within 16-lane halves, ds_add_f32
// into LDS, then one global_atomic_add_f32 per row per block.
// -------------------------------------------------------------------------
__global__ __launch_bounds__(256) void lorentz_expscores_kernel(
    const float* __restrict__ x, const float* __restrict__ temperature,
    float* __restrict__ attn, float* __restrict__ rowsum)
{
    __shared__ float Xi[64][KPAD];   // A-side (time component negated)
    __shared__ float Xj[64][KPAD];   // B-side
    __shared__ float rsum[64];
    const int b  = blockIdx.z;
    const int i0 = blockIdx.y * 64;
    const int j0 = blockIdx.x * 64;
    const int t  = threadIdx.x;
    const float inv_temp = 1.0f / (temperature[0] + TEMP_EPS);

    if (t < 64) rsum[t] = 0.0f;
    const float* xb = x + (size_t)b * SS * DD;
    for (int idx = t; idx < 64 * KPAD; idx += 256) {
        const int r = idx / KPAD, d = idx - r * KPAD;
        float vi = 0.0f, vj = 0.0f;
        if (d < DD) {
            vi = xb[(size_t)(i0 + r) * DD + d];
            vj = xb[(size_t)(j0 + r) * DD + d];
        }
        Xi[r][d] = (d == 0) ? -vi : vi;   // Lorentz signature on A operand
        Xj[r][d] = vj;
    }
    __syncthreads();

    const int lane  = t & 31;
    const int wave  = t >> 5;
    const int m0    = (wave >> 2) * 32;   // rows m0..m0+31 for this wave
    const int n0    = (wave & 3) * 16;    // col tile
    const int lrow  = lane & 15;
    const int khalf = (lane >> 4) << 1;   // lanes 16-31 hold K+2,K+3

    v8f acc0 = {};
    v8f acc1 = {};
    for (int kk = 0; kk < KPAD; kk += 4) {
        const int kb = kk + khalf;
        v2f a0, a1, bf;
        a0.x = Xi[m0 + lrow][kb];        a0.y = Xi[m0 + lrow][kb + 1];
        a1.x = Xi[m0 + 16 + lrow][kb];   a1.y = Xi[m0 + 16 + lrow][kb + 1];
        bf.x = Xj[n0 + lrow][kb];        bf.y = Xj[n0 + lrow][kb + 1];
        acc0 = __builtin_amdgcn_wmma_f32_16x16x4_f32(
                   false, a0, false, bf, (short)0, acc0, false, false);
        acc1 = __builtin_amdgcn_wmma_f32_16x16x4_f32(
                   false, a1, false, bf, (short)0, acc1, false, false);
    }

    // Epilogue. C/D layout: VGPR v, lanes 0-15 -> M=v, 16-31 -> M=v+8; N=lane&15
    const int N    = j0 + n0 + lrow;
    const int Mb   = (lane >> 4) * 8;
    float* ab = attn + (size_t)b * SS * SS;
#pragma unroll
    for (int v = 0; v < 8; ++v) {
        float u0 = fminf(fmaxf(-acc0[v], ACOSH_MIN), ACOSH_MAX);
        float u1 = fminf(fmaxf(-acc1[v], ACOSH_MIN), ACOSH_MAX);
        float d0 = __logf(u0 + __builtin_sqrtf(__builtin_fmaf(u0, u0, -1.0f)));
        float d1 = __logf(u1 + __builtin_sqrtf(__builtin_fmaf(u1, u1, -1.0f)));
        float p0 = __expf(-d0 * inv_temp);            // bounded: (0, 1]
        float p1 = __expf(-d1 * inv_temp);
        ab[(size_t)(i0 + m0 + Mb + v) * SS + N]      = p0;
        ab[(size_t)(i0 + m0 + 16 + Mb + v) * SS + N] = p1;
        // cross-lane sum within each 16-lane half (rows differ per half)
        float s0 = p0, s1 = p1;
#pragma unroll
        for (int mask = 1; mask < 16; mask <<= 1) {
            s0 += __shfl_xor(s0, mask, 32);
            s1 += __shfl_xor(s1, mask, 32);
        }
        if ((lane & 15) == 0) {
            atomicAdd(&rsum[m0 + Mb + v], s0);        // ds_add_f32
            atomicAdd(&rsum[m0 + 16 + Mb + v], s1);
        }
    }
    __syncthreads();
    if (t < 64) atomicAdd(&rowsum[(size_t)b * SS + i0 + t], rsum[t]);
}

// -------------------------------------------------------------------------
// Kernel 3: output = attention @ values, fused with normalization.
// 32-row stripe x N=80 (D padded to 5 N-tiles), 320 threads = 10 waves =
// 2(M) x 5(N) 16x16 fp32-WMMA subtiles. While staging each 32x64 A tile
// (b128 loads) it multiplies by 1/rowsum and writes the NORMALIZED
// attention back in place (rows owned exclusively per block), so the
// attention matrix is read once and written once here.
// -------------------------------------------------------------------------
constexpr int KC = 64;   // K chunk
constexpr int NP = 80;   // D padded to 5*16 (NP%64=16 -> conflict-free B reads)

__global__ __launch_bounds__(320) void attn_values_kernel(
    float* __restrict__ attn, const float* __restrict__ values,
    const float* __restrict__ rowsum, float* __restrict__ out)
{
    __shared__ float As[32][KC + 4];  // stride 68: A-frag reads conflict-free
    __shared__ float Vs[KC][NP];
    __shared__ float inv32[32];
    const int b  = blockIdx.y;
    const int i0 = blockIdx.x * 32;
    const int t  = threadIdx.x;
    float* arow     = attn + ((size_t)b * SS + i0) * SS;
    const float* vb = values + (size_t)b * SS * DD;

    if (t < 32) inv32[t] = 1.0f / rowsum[(size_t)b * SS + i0 + t];
    __syncthreads();

    const int lane  = t & 31;
    const int wave  = t >> 5;
    const int m0    = (wave / 5) * 16;
    const int n0    = (wave % 5) * 16;
    const int lrow  = lane & 15;
    const int khalf = (lane >> 4) << 1;

    v8f acc = {};
    for (int k0 = 0; k0 < SS; k0 += KC) {
        if (k0 + KC < SS) {   // gfx1250 global_prefetch_b8 on next value chunk
            __builtin_prefetch(vb + (size_t)(k0 + KC) * DD, 0, 0);
        }
        // stage + normalize + in-place write-back, 16B granularity
        for (int idx = t; idx < 32 * (KC / 4); idx += 320) {
            const int r  = idx >> 4;
            const int c4 = (idx & 15) << 2;
            float4 pv = *(const float4*)(arow + (size_t)r * SS + k0 + c4);
            const float s = inv32[r];
            pv.x *= s; pv.y *= s; pv.z *= s; pv.w *= s;
            *(float4*)(arow + (size_t)r * SS + k0 + c4) = pv;  // final attention
            As[r][c4 + 0] = pv.x; As[r][c4 + 1] = pv.y;
            As[r][c4 + 2] = pv.z; As[r][c4 + 3] = pv.w;
        }
        for (int idx = t; idx < KC * NP; idx += 320) {
            const int r = idx / NP, c = idx - r * NP;
            Vs[r][c] = (c < DD) ? vb[(size_t)(k0 + r) * DD + c] : 0.0f;
        }
        __syncthreads();
#pragma unroll
        for (int kk = 0; kk < KC; kk += 4) {
            const int kb = kk + khalf;
            v2f a, bf;
            a.x  = As[m0 + lrow][kb];   a.y  = As[m0 + lrow][kb + 1];
            bf.x = Vs[kb][n0 + lrow];   bf.y = Vs[kb + 1][n0 + lrow];
            acc = __builtin_amdgcn_wmma_f32_16x16x4_f32(
                      false, a, false, bf, (short)0, acc, false, false);
        }
        __syncthreads();
    }

    const int N  = n0 + lrow;
    const int Mb = (lane >> 4) * 8;
    if (N < DD) {
#pragma unroll
        for (int v = 0; v < 8; ++v) {
            out[((size_t)b * SS + i0 + m0 + Mb + v) * DD + N] = acc[v];
        }
    }
}

// -------------------------------------------------------------------------
extern "C" void kernel_launch(void* const* d_in, const int* in_sizes, int n_in,
                              void* d_out, int out_size, void* d_ws, size_t ws_size,
                              hipStream_t stream) {
    (void)in_sizes; (void)n_in; (void)out_size; (void)ws_size;

    const float* x    = (const float*)d_in[0];
    const float* vals = (const float*)d_in[1];
    const float* temp = (const float*)d_in[2];

    float* out    = (float*)d_out;                    // B*S*D
    float* attn   = out + (size_t)BB * SS * DD;       // B*S*S
    float* rowsum = (float*)d_ws;                     // B*S floats (64 KB)

    zero_rowsums_kernel<<<(BB * SS + 255) / 256, 256, 0, stream>>>(rowsum);

    dim3 g1(SS / 64, SS / 64, BB);
    lorentz_expscores_kernel<<<g1, 256, 0, stream>>>(x, temp, attn, rowsum);

    dim3 g3(SS / 32, BB);
    attn_values_kernel<<<g3, 320, 0, stream>>>(attn, vals, rowsum, out);
}